// Attention_34986803593304
// MI455X (gfx1250) — compile-verified
//
#include <hip/hip_runtime.h>
#include <hip/hip_bf16.h>

typedef float v2f __attribute__((ext_vector_type(2)));
typedef float v4f __attribute__((ext_vector_type(4)));
typedef float v8f __attribute__((ext_vector_type(8)));

#define DICT   20
#define NOUT   200      // NUM_HEADS * DICT
#define NPAD   208      // 13 tiles of 16
#define NTILES 13
#define BATCH  262144
#define WPB    4        // waves per block
#define LDS_STRIDE 212  // 16-row tile stride; 8*212 % 64 == 32 -> conflict-free C writes

// Kernel 1: Weff[n][k] = gamma*W3[n][k] + W4[n][k], zero-padded to 208 rows.
// 16.6 KB -> resident in L1/L2 for the whole main kernel (kept regular-temporal).
__global__ void prep_weff(const float* __restrict__ W3,
                          const float* __restrict__ W4,
                          const float* __restrict__ gamma,
                          float* __restrict__ Wp) {
    const float g = gamma[0];
    for (int i = threadIdx.x; i < NPAD * DICT; i += blockDim.x) {
        const int n = i / DICT;
        const int k = i % DICT;
        Wp[i] = (n < NOUT) ? fmaf(g, W3[n * DICT + k], W4[n * DICT + k]) : 0.0f;
    }
}

// Kernel 2: out = relu(x @ Weff^T) via V_WMMA_F32_16X16X4_F32.
// One wave per 16-row batch tile. C tiles staged in per-wave LDS, then written
// back as 25 fully-coalesced non-temporal b128 stores (tile is contiguous in out).
__global__ __launch_bounds__(128) void fused_attn_gemm(
    const float* __restrict__ x,
    const float* __restrict__ Wp,
    float* __restrict__ out) {

    __shared__ float stage[WPB * 16 * LDS_STRIDE];   // 54,272 B

    const int lane = threadIdx.x & 31;
    const int wave = threadIdx.x >> 5;
    float* my = stage + wave * (16 * LDS_STRIDE);    // per-wave private region

    const int tile = blockIdx.x * WPB + wave;        // 16-row batch tile
    const int row0 = tile * 16;

    // fp32 16x4 A layout: lanes 0-15 -> M=0..15, K={0,1}; lanes 16-31 -> K={2,3}
    const int mrow = lane & 15;
    const int kb   = (lane >> 4) * 2;

    // x is read exactly once -> non-temporal loads keep it out of L2 residency.
    v2f a[5];
    const float* xr = x + (size_t)(row0 + mrow) * DICT;
#pragma unroll
    for (int kc = 0; kc < 5; ++kc) {
        a[kc] = __builtin_nontemporal_load((const v2f*)(xr + kc * 4 + kb));
    }

    // B (4x16) mirrored layout: lane%16 -> N, lane-half + VGPR slot -> K.
    const int ncol  = lane & 15;
    const int mbase = (lane >> 4) << 3;              // C layout: lanes 16-31 hold M=8..15

#pragma unroll 1                                     // keep one live accumulator
    for (int nt = 0; nt < NTILES; ++nt) {
        v8f c = {};
#pragma unroll
        for (int kc = 0; kc < 5; ++kc) {
            v2f b = *(const v2f*)(Wp + (nt * 16 + ncol) * DICT + kc * 4 + kb);
            // (neg_a, A, neg_b, B, c_mod, C, reuse_a, reuse_b)
            c = __builtin_amdgcn_wmma_f32_16x16x4_f32(
                    false, a[kc], false, b, (short)0, c, false, false);
        }
        // ReLU + stage to LDS (conflict-free: half-waves hit disjoint banks)
#pragma unroll
        for (int i = 0; i < 8; ++i) {
            my[(mbase + i) * LDS_STRIDE + nt * 16 + ncol] = fmaxf(c[i], 0.0f);
        }
    }

    // Coalesced writeback: the wave's 16x200 tile is one contiguous 12.8 KB
    // span of out. 25 chunks x (32 lanes x float4) = 512 B per store instr.
    // Same-wave DS ops are in-order, so no barrier is needed. Unroll by 5 so
    // the DS loads clause together instead of serializing on s_wait_dscnt.
    float* obase = out + (size_t)row0 * NOUT;
#pragma unroll 5
    for (int t = 0; t < 25; ++t) {
        const int g0 = (t * 32 + lane) * 4;          // 0..3196, multiple of 4
        const int m  = g0 / NOUT;                    // 200 % 4 == 0 -> no row straddle
        const int cc = g0 - m * NOUT;
        const v4f v = *(const v4f*)(my + m * LDS_STRIDE + cc);   // 16B-aligned
        // Streaming 210 MB output: non-temporal keeps it from thrashing L2.
        __builtin_nontemporal_store(v, (v4f*)(obase + g0));      // 16B-aligned
    }
}

extern "C" void kernel_launch(void* const* d_in, const int* in_sizes, int n_in,
                              void* d_out, int out_size, void* d_ws, size_t ws_size,
                              hipStream_t stream) {
    // setup_inputs order: x, W1, W2, W3, W4, gamma.  W1/W2 are dead (softmax over
    // a length-1 axis is identity, so res == x @ W3^T exactly).
    const float* x     = (const float*)d_in[0];
    const float* W3    = (const float*)d_in[3];
    const float* W4    = (const float*)d_in[4];
    const float* gamma = (const float*)d_in[5];
    float*       Wp    = (float*)d_ws;               // 208*20*4 = 16,640 bytes
    float*       out   = (float*)d_out;

    prep_weff<<<1, 256, 0, stream>>>(W3, W4, gamma, Wp);

    const int nblocks = BATCH / 16 / WPB;            // 4096
    fused_attn_gemm<<<nblocks, 128, 0, stream>>>(x, Wp, out);
}